// SilenceREIN_72567767433445
// MI455X (gfx1250) — compile-verified
//
#include <hip/hip_runtime.h>
#include <math.h>

typedef __attribute__((ext_vector_type(2))) float v2f;
typedef __attribute__((ext_vector_type(8))) float v8f;

#define N_NODES   131072
#define N_GRAPHS  2048
#define N_EDGES   2097152

static __device__ __forceinline__ v8f wmma4(v2f a, v2f b, v8f c) {
  // D(16x16,f32) = A(16x4,f32) x B(4x16,f32) + C   -- full f32 precision
  return __builtin_amdgcn_wmma_f32_16x16x4_f32(false, a, false, b, (short)0, c, false, false);
}

__global__ void zero_f32(float* __restrict__ p, long n) {
  long i = (long)blockIdx.x * blockDim.x + threadIdx.x;
  if (i < n) p[i] = 0.0f;
}

__global__ void edge_scatter(const int* __restrict__ ei, const float* __restrict__ x1,
                             float* __restrict__ msg, float* __restrict__ deg) {
  int e = blockIdx.x * blockDim.x + threadIdx.x;
  if (e >= N_EDGES) return;
  int src = ei[e];
  int dst = ei[N_EDGES + e];
  const float* xs = x1 + (long)src * 10;
  float* md = msg + (long)dst * 10;
#pragma unroll
  for (int k = 0; k < 10; ++k) atomicAdd(&md[k], xs[k]);
  atomicAdd(&deg[dst], 1.0f);
}

// Fused: mean-agg division, SAGE linear (h = mean@Wl^T + bl + x1@Wr^T), graph add-pool.
// One block (128 thr) per graph; writes gp as [100][N_GRAPHS] (feature-major).
__global__ void sage_pool(const float* __restrict__ x1, const float* __restrict__ msg,
                          const float* __restrict__ deg,
                          const float* __restrict__ Wl, const float* __restrict__ bl,
                          const float* __restrict__ Wr, float* __restrict__ gp) {
  __shared__ float sx[64][10];
  __shared__ float sm[64][10];
  int graph = blockIdx.x;
  int t = threadIdx.x;
  for (int idx = t; idx < 640; idx += blockDim.x) {
    int node = idx / 10, k = idx % 10;
    long gn = (long)graph * 64 + node;
    sx[node][k] = x1[gn * 10 + k];
    float d = deg[gn];
    sm[node][k] = msg[gn * 10 + k] / fmaxf(d, 1.0f);
  }
  __syncthreads();
  if (t < 100) {
    float wl[10], wr[10];
#pragma unroll
    for (int k = 0; k < 10; ++k) { wl[k] = Wl[t * 10 + k]; wr[k] = Wr[t * 10 + k]; }
    float acc = 64.0f * bl[t];
    for (int n = 0; n < 64; ++n) {
#pragma unroll
      for (int k = 0; k < 10; ++k)
        acc += sm[n][k] * wl[k] + sx[n][k] * wr[k];
    }
    gp[t * N_GRAPHS + graph] = acc;
  }
}

// Conv1d(k=3, pad=1) as GEMM, fused bias + leaky_relu(0.01) + maxpool3(stride 3, pad PPAD).
// Block = 7 waves (one per 16-row M-tile covering 100 out channels), one pooled 16-tile
// in N per block. Input tile (50 cols, all in-channels) staged zero-padded in LDS so the
// WMMA inner loop has NO predicated loads: B frags come from ds_load, A frags from
// clamped global addresses times a 0/1 lane mask.
template<int IC, int LIN, int LPOOL, int PPAD>
__global__ __launch_bounds__(224) void conv_pool(const float* __restrict__ in,
                                                 const float* __restrict__ w,
                                                 const float* __restrict__ bias,
                                                 float* __restrict__ out) {
  constexpr int IC3 = IC * 3;
  constexpr int KP  = (IC3 + 3) & ~3;
  constexpr int ICP = (KP + 2) / 3;      // staged rows incl. zero row for K padding
  __shared__ float sin_[ICP * 52];
  __shared__ float sbuf[7 * 16 * 48];
  const int tid  = threadIdx.x;
  const int wave = tid >> 5;
  const int lane = tid & 31;
  const int half = lane >> 4, nl = lane & 15;
  const int pt    = blockIdx.x;
  const int graph = blockIdx.y;
  const float* gin = in + (long)graph * IC * LIN;
  const int lbase = pt * 48 - PPAD;      // first conv position of this tile

  // Stage input columns [lbase-1, lbase+49), zero-padded (incl. extra zero rows).
  for (int idx = tid; idx < ICP * 50; idx += 224) {
    int i = idx / 50, c = idx - i * 50;
    int gpos = lbase - 1 + c;
    float v = 0.0f;
    if (i < IC && gpos >= 0 && gpos < LIN) v = gin[i * LIN + gpos];
    sin_[i * 52 + c] = v;
  }
  __syncthreads();

  const int   mt   = wave;
  const int   m    = mt * 16 + nl;
  const float msel = (m < 100) ? 1.0f : 0.0f;
  const float* wrow = w + ((m < 100) ? m : 99) * IC3;   // always-valid address

  v8f c0 = {}, c1 = {}, c2 = {};
  for (int k0 = 0; k0 < KP; k0 += 4) {
    int kkx = k0 + 2 * half;
    int kky = kkx + 1;
    float aselx = msel, asely = msel;
    int kwx = kkx, kwy = kky;
    if constexpr (KP != IC3) {           // only conv1 has a K tail
      if (kkx >= IC3) { aselx = 0.0f; kwx = 0; }
      if (kky >= IC3) { asely = 0.0f; kwy = 0; }
    }
    v2f a;
    a.x = wrow[kwx] * aselx;
    a.y = wrow[kwy] * asely;
    int ix = kkx / 3, dx = kkx - 3 * ix; // kk >= IC3 lands in the zero LDS row
    int iy = kky / 3, dy = kky - 3 * iy;
    const float* bx = &sin_[ix * 52 + dx];
    const float* by = &sin_[iy * 52 + dy];
#pragma unroll
    for (int j = 0; j < 3; ++j) {
      int col = j * 16 + nl;
      v2f b;
      b.x = bx[col];
      b.y = by[col];
      if (j == 0)      c0 = wmma4(a, b, c0);
      else if (j == 1) c1 = wmma4(a, b, c1);
      else             c2 = wmma4(a, b, c2);
    }
  }

  // bias + leaky relu -> per-wave LDS strip (16 rows x 48 conv positions)
  float* sb = &sbuf[wave * 768];
#pragma unroll
  for (int v = 0; v < 8; ++v) {
    int row = v + 8 * half;
    int mm  = mt * 16 + row;
    float bv = bias[(mm < 100) ? mm : 99];   // rows >=100 never written out
    float vals[3] = {c0[v], c1[v], c2[v]};
#pragma unroll
    for (int j = 0; j < 3; ++j) {
      int l = lbase + j * 16 + nl;
      float x = vals[j] + bv;
      x = (x > 0.0f) ? x : 0.01f * x;
      if (l < 0 || l >= LIN) x = -1e30f;     // -inf padding for maxpool
      sb[row * 48 + j * 16 + nl] = x;
    }
  }
  __syncthreads();
#pragma unroll
  for (int t = 0; t < 8; ++t) {
    int idx = lane * 8 + t;                  // 256 pooled elements: 16 rows x 16 p
    int row = idx >> 4;
    int p   = idx & 15;
    const float* r = &sb[row * 48 + p * 3];
    float x = fmaxf(fmaxf(r[0], r[1]), r[2]);
    int mm   = mt * 16 + row;
    int pout = pt * 16 + p;
    if (mm < 100 && pout < LPOOL)
      out[((long)graph * 100 + mm) * LPOOL + pout] = x;
  }
}

// Concat [g ; cnn-flat] feature, BatchNorm with batch statistics, write feature-major.
__global__ void bn_concat(const float* __restrict__ gp, const float* __restrict__ y4,
                          const float* __restrict__ gam, const float* __restrict__ bet,
                          float* __restrict__ zn) {
  __shared__ float s1[256], s2[256];
  int f = blockIdx.x;      // 0..899
  int t = threadIdx.x;     // 256 threads, 8 graphs each
  float vals[8];
  float s = 0.0f, sq = 0.0f;
#pragma unroll
  for (int i = 0; i < 8; ++i) {
    int g = t + i * 256;
    float v = (f < 100) ? gp[f * N_GRAPHS + g] : y4[(long)g * 800 + (f - 100)];
    vals[i] = v; s += v; sq += v * v;
  }
  s1[t] = s; s2[t] = sq;
  __syncthreads();
  for (int o = 128; o > 0; o >>= 1) {
    if (t < o) { s1[t] += s1[t + o]; s2[t] += s2[t + o]; }
    __syncthreads();
  }
  float mu  = s1[0] * (1.0f / N_GRAPHS);
  float var = s2[0] * (1.0f / N_GRAPHS) - mu * mu;
  float inv = rsqrtf(var + 1e-5f);
  float sc = gam[f] * inv;
  float sh = bet[f] - mu * sc;
#pragma unroll
  for (int i = 0; i < 8; ++i) {
    int g = t + i * 256;
    zn[(long)f * N_GRAPHS + g] = vals[i] * sc + sh;
  }
}

// D[M,N] = relu(A[M,K] x B[K,N] + bias). One wave per 16x64 tile: each A fragment is
// reused by 4 WMMAs; B loads are 64-wide contiguous per k row.
template<int M, int N, int K, bool RELU>
__global__ __launch_bounds__(32) void gemm_bias(const float* __restrict__ A,
                                                const float* __restrict__ B,
                                                const float* __restrict__ bias,
                                                float* __restrict__ D) {
  constexpr int NB = N / 64;
  const int lane = threadIdx.x;
  const int half = lane >> 4, nl = lane & 15;
  const int mt = blockIdx.x / NB;
  const int nb = blockIdx.x % NB;
  const int m  = mt * 16 + nl;
  const int n0 = nb * 64;
  v8f c0 = {}, c1 = {}, c2 = {}, c3 = {};
  for (int k0 = 0; k0 < K; k0 += 4) {
    int kk = k0 + 2 * half;
    v2f a;
    a.x = A[m * K + kk];
    a.y = A[m * K + kk + 1];
    const float* Bp = B + (long)kk * N + n0 + nl;
    const float* Bq = B + (long)(kk + 1) * N + n0 + nl;
#pragma unroll
    for (int j = 0; j < 4; ++j) {
      v2f b;
      b.x = Bp[j * 16];
      b.y = Bq[j * 16];
      if (j == 0)      c0 = wmma4(a, b, c0);
      else if (j == 1) c1 = wmma4(a, b, c1);
      else if (j == 2) c2 = wmma4(a, b, c2);
      else             c3 = wmma4(a, b, c3);
    }
  }
  v8f acc[4] = {c0, c1, c2, c3};
#pragma unroll
  for (int j = 0; j < 4; ++j) {
#pragma unroll
    for (int v = 0; v < 8; ++v) {
      int mm = mt * 16 + v + 8 * half;
      float x = acc[j][v] + bias[mm];
      if (RELU) x = fmaxf(x, 0.0f);
      D[(long)mm * N + n0 + j * 16 + nl] = x;
    }
  }
}

__global__ void fc3_kernel(const float* __restrict__ z3, const float* __restrict__ W,
                           const float* __restrict__ b, float* __restrict__ out) {
  int id = blockIdx.x * blockDim.x + threadIdx.x;
  if (id >= N_GRAPHS * 2) return;
  int g = id >> 1, o = id & 1;
  float acc = b[o];
#pragma unroll
  for (int k = 0; k < 64; ++k) acc += W[o * 64 + k] * z3[k * N_GRAPHS + g];
  out[g * 2 + o] = acc;
}

extern "C" void kernel_launch(void* const* d_in, const int* in_sizes, int n_in,
                              void* d_out, int out_size, void* d_ws, size_t ws_size,
                              hipStream_t stream) {
  (void)in_sizes; (void)n_in; (void)out_size; (void)ws_size;
  const float* x1  = (const float*)d_in[0];
  const float* x2  = (const float*)d_in[1];
  const int*   ei  = (const int*)d_in[2];
  /* d_in[3] = batch: contiguous node//64, not needed */
  const float* Wl  = (const float*)d_in[4];
  const float* bl  = (const float*)d_in[5];
  const float* Wr  = (const float*)d_in[6];
  const float* cw1 = (const float*)d_in[7];
  const float* cb1 = (const float*)d_in[8];
  const float* cw2 = (const float*)d_in[9];
  const float* cb2 = (const float*)d_in[10];
  const float* cw3 = (const float*)d_in[11];
  const float* cb3 = (const float*)d_in[12];
  const float* cw4 = (const float*)d_in[13];
  const float* cb4 = (const float*)d_in[14];
  const float* bng = (const float*)d_in[15];
  const float* bnb = (const float*)d_in[16];
  const float* W1  = (const float*)d_in[17];
  const float* b1  = (const float*)d_in[18];
  const float* W2  = (const float*)d_in[19];
  const float* b2  = (const float*)d_in[20];
  const float* W3  = (const float*)d_in[21];
  const float* b3  = (const float*)d_in[22];

  float* ws   = (float*)d_ws;
  float* msg  = ws;                      // [131072,10]
  float* deg  = msg  + 1310720;          // [131072]
  float* gp   = deg  + 131072;           // [100,2048]
  float* buf2 = gp   + 204800;           // [2048,100,67]
  float* buf1 = buf2 + 13721600;         // [2048,100,200] (largest; freed after L2)
  float* buf3 = buf1;                    // alias into buf1 region: [2048,100,23]
  float* buf4 = buf3 + 4710400;          // [2048,100,8]
  float* zn   = buf4 + 1638400;          // [900,2048]
  float* z2   = zn   + 1843200;          // [256,2048]
  float* z3   = z2   + 524288;           // [64,2048]

  long nz = 1310720 + 131072;
  zero_f32<<<dim3((unsigned)((nz + 255) / 256)), dim3(256), 0, stream>>>(msg, nz);
  edge_scatter<<<dim3((N_EDGES + 255) / 256), dim3(256), 0, stream>>>(ei, x1, msg, deg);
  sage_pool<<<dim3(N_GRAPHS), dim3(128), 0, stream>>>(x1, msg, deg, Wl, bl, Wr, gp);

  conv_pool<21, 600, 200, 0><<<dim3(13, N_GRAPHS), dim3(224), 0, stream>>>(x2,  cw1, cb1, buf1);
  conv_pool<100, 200, 67, 1><<<dim3(5,  N_GRAPHS), dim3(224), 0, stream>>>(buf1, cw2, cb2, buf2);
  conv_pool<100, 67,  23, 1><<<dim3(2,  N_GRAPHS), dim3(224), 0, stream>>>(buf2, cw3, cb3, buf3);
  conv_pool<100, 23,  8,  1><<<dim3(1,  N_GRAPHS), dim3(224), 0, stream>>>(buf3, cw4, cb4, buf4);

  bn_concat<<<dim3(900), dim3(256), 0, stream>>>(gp, buf4, bng, bnb, zn);
  gemm_bias<256, 2048, 900, true><<<dim3(16 * 32), dim3(32), 0, stream>>>(W1, zn, b1, z2);
  gemm_bias<64,  2048, 256, true><<<dim3(4 * 32),  dim3(32), 0, stream>>>(W2, z2, b2, z3);
  fc3_kernel<<<dim3((N_GRAPHS * 2 + 255) / 256), dim3(256), 0, stream>>>(z3, W3, b3, (float*)d_out);
}